// BiLSTM_CRF_4964982194587
// MI455X (gfx1250) — compile-verified
//
#include <hip/hip_runtime.h>
#include <hip/hip_bf16.h>

#define VOCAB   100000
#define K_TAGS  17
#define E_DIM   100
#define H_DIM   128
#define HD      64
#define B_SZ    256
#define T_SZ    512
#define G4      256          // 4*HD gates per direction
#define E_PAD   108          // E_DIM pad: 432B rows (16B aligned, bank-clean)
#define HD_PAD  66           // HD + 2
#define H3_PAD  132          // H_DIM pad: 528B rows (16B aligned)
#define GX_PAD  264          // G4 pad: 1056B rows (16B aligned)

typedef __attribute__((ext_vector_type(2))) float v2f;
typedef __attribute__((ext_vector_type(8))) float v8f;

static __device__ __forceinline__ v8f wmma_f32(v2f a, v2f b, v8f c) {
    // D = A(16x4) * B(4x16) + C, fp32 WMMA (wave32)
    return __builtin_amdgcn_wmma_f32_16x16x4_f32(
        /*neg_a=*/false, a, /*neg_b=*/false, b,
        /*c_mod=*/(short)0, c, /*reuse_a=*/false, /*reuse_b=*/false);
}

static __device__ __forceinline__ float sigf(float x) {
    return 1.0f / (1.0f + __expf(-x));
}

// ---- CDNA5 async global->LDS (ASYNCcnt-tracked), inline asm for portability
static __device__ __forceinline__ void async_b128(uint32_t lds, const void* g) {
    asm volatile("global_load_async_to_lds_b128 %0, %1, off"
                 :: "v"(lds), "v"((uint64_t)(uintptr_t)g) : "memory");
}
static __device__ __forceinline__ void async_b64(uint32_t lds, const void* g) {
    asm volatile("global_load_async_to_lds_b64 %0, %1, off"
                 :: "v"(lds), "v"((uint64_t)(uintptr_t)g) : "memory");
}
#define WAIT_ASYNC(n) asm volatile("s_wait_asynccnt " #n ::: "memory")
#define LDSA(p) ((uint32_t)(uintptr_t)(p))

// ---------------------------------------------------------------------------
// Kernel 1: embedding gather fused with input projection for BOTH directions.
//   gx[dir][m][g] = sum_e emb[sent[m]][e] * w_ih[g][e] + b_ih[g]
// Grid: (BT/16, 4). Block: 256 (8 waves). Each wave: one 16x16 C tile.
// A/B tiles staged with async global->LDS loads.
// ---------------------------------------------------------------------------
__global__ void k_input_gemm(const int* __restrict__ sent,
                             const float* __restrict__ emb,
                             const float* __restrict__ w_ih_f,
                             const float* __restrict__ b_ih_f,
                             const float* __restrict__ w_ih_b,
                             const float* __restrict__ b_ih_b,
                             float* __restrict__ gx)
{
    __shared__ __align__(16) float As[16][E_PAD];    // x tile   (M x K)
    __shared__ __align__(16) float Bs[128][E_PAD];   // weights  (N x K)

    const int tid  = threadIdx.x;
    const int m0   = blockIdx.x * 16;
    const int nblk = blockIdx.y * 128;

    // A: embedding gather. Rows are 400B offsets -> 8B aligned -> B64 chunks.
    for (int idx = tid; idx < 16 * 50; idx += 256) {
        int r = idx / 50, c = idx - r * 50, e = 2 * c;
        int s = sent[m0 + r];
        async_b64(LDSA(&As[r][e]), emb + (size_t)s * E_DIM + e);
    }
    // B: [w_ih_f ; w_ih_b] rows, 400B row offsets -> 16B aligned -> B128.
    for (int idx = tid; idx < 128 * 25; idx += 256) {
        int gl = idx / 25, c = idx - gl * 25, e = 4 * c;
        int g = nblk + gl;
        const float* src = (g < G4) ? (w_ih_f + g * E_DIM + e)
                                    : (w_ih_b + (g - G4) * E_DIM + e);
        async_b128(LDSA(&Bs[gl][e]), src);
    }
    WAIT_ASYNC(0);
    __syncthreads();

    const int wv = tid >> 5, lane = tid & 31;
    const int half = lane >> 4, lrow = lane & 15;
    const int nloc = wv * 16;

    v8f c = {};
    #pragma unroll 5
    for (int kk = 0; kk < E_DIM; kk += 4) {   // 25 WMMA steps (K=100)
        int k = kk + half * 2;
        v2f a, b;
        a[0] = As[lrow][k];         a[1] = As[lrow][k + 1];
        b[0] = Bs[nloc + lrow][k];  b[1] = Bs[nloc + lrow][k + 1];
        c = wmma_f32(a, b, c);
    }

    const int ng  = nblk + nloc + lrow;            // 0..511
    const int dir = ng >> 8;
    const int g   = ng & 255;
    const float bias = (dir == 0) ? b_ih_f[g] : b_ih_b[g];
    const size_t base = (size_t)dir * (B_SZ * T_SZ) * G4;
    #pragma unroll
    for (int r = 0; r < 8; ++r) {
        int row = m0 + r + half * 8;
        gx[base + (size_t)row * G4 + g] = c[r] + bias;
    }
}

// ---------------------------------------------------------------------------
// Kernel 2: LSTM recurrence. One direction per blockIdx.y, 16 batch rows per
// block, all T steps. W_hh lives in REGISTERS in WMMA B-layout; h lives in
// LDS as the WMMA A-matrix. gx[t] tiles are double-buffered in LDS and
// prefetched one step ahead with async global->LDS, overlapping the WMMAs.
// Grid: (16, 2). Block: 256 (8 waves). Each wave owns 2 N-tiles (32 gates).
// ---------------------------------------------------------------------------
__global__ void k_lstm(const float* __restrict__ gx,
                       const float* __restrict__ w_hh_f,
                       const float* __restrict__ b_hh_f,
                       const float* __restrict__ w_hh_b,
                       const float* __restrict__ b_hh_b,
                       float* __restrict__ hout)
{
    __shared__ __align__(16) float Hs[16][HD_PAD];      // h state (A-matrix)
    __shared__ __align__(16) float Gs[16][258];         // gate pre-activations
    __shared__ __align__(16) float Gxs[2][16][GX_PAD];  // gx[t] double buffer

    const int dir = blockIdx.y;
    const int b0  = blockIdx.x * 16;
    const int tid = threadIdx.x;
    const float* whh = dir ? w_hh_b : w_hh_f;
    const float* bhh = dir ? b_hh_b : b_hh_f;

    const int wv = tid >> 5, lane = tid & 31;
    const int half = lane >> 4, lrow = lane & 15;
    const int n0 = wv * 32;

    // Preload this wave's W_hh slab into registers in WMMA B layout.
    v2f wb0[16], wb1[16];
    #pragma unroll
    for (int kk = 0; kk < 16; ++kk) {
        int k = kk * 4 + half * 2;
        wb0[kk][0] = whh[(n0 + lrow) * HD + k];
        wb0[kk][1] = whh[(n0 + lrow) * HD + k + 1];
        wb1[kk][0] = whh[(n0 + 16 + lrow) * HD + k];
        wb1[kk][1] = whh[(n0 + 16 + lrow) * HD + k + 1];
    }
    const float bhA = bhh[n0 + lrow];
    const float bhB = bhh[n0 + 16 + lrow];

    // Zero h state; each thread owns 4 (row,h) cell states in registers.
    for (int idx = tid; idx < 16 * HD; idx += 256) Hs[idx >> 6][idx & 63] = 0.f;
    float cst[4];
    int prow[4], ph[4];
    #pragma unroll
    for (int j = 0; j < 4; ++j) {
        int p = tid + 256 * j;
        prow[j] = p >> 6; ph[j] = p & 63; cst[j] = 0.f;
    }

    const size_t gxbase = (size_t)dir * (B_SZ * T_SZ) * G4;
    const size_t hbase  = (size_t)dir * (B_SZ * T_SZ) * HD;

    // Prefetch gx tile for step 0 (4 B128 chunks per thread, 1024 total).
    {
        const int t0 = dir ? (T_SZ - 1) : 0;
        #pragma unroll
        for (int j = 0; j < 4; ++j) {
            int ch = tid + 256 * j;             // 16 rows x 64 col-groups
            int row = ch >> 6, cg = ch & 63;
            async_b128(LDSA(&Gxs[0][row][cg * 4]),
                       gx + gxbase + ((size_t)(b0 + row) * T_SZ + t0) * G4 + cg * 4);
        }
    }
    __syncthreads();

    for (int step = 0; step < T_SZ; ++step) {
        const int t  = dir ? (T_SZ - 1 - step) : step;
        const int sn = (step + 1 < T_SZ) ? (step + 1) : step;   // clamp (dummy)
        const int tn = dir ? (T_SZ - 1 - sn) : sn;
        const int cb = step & 1, nb = (step + 1) & 1;

        // Issue async prefetch of next step's gx tile (overlaps the WMMAs).
        #pragma unroll
        for (int j = 0; j < 4; ++j) {
            int ch = tid + 256 * j;
            int row = ch >> 6, cg = ch & 63;
            async_b128(LDSA(&Gxs[nb][row][cg * 4]),
                       gx + gxbase + ((size_t)(b0 + row) * T_SZ + tn) * G4 + cg * 4);
        }

        // h @ Whh^T for this wave's two 16-wide gate tiles.
        v8f c0 = {}, c1 = {};
        #pragma unroll
        for (int kk = 0; kk < 16; ++kk) {     // K = 64
            int k = kk * 4 + half * 2;
            v2f a;
            a[0] = Hs[lrow][k]; a[1] = Hs[lrow][k + 1];
            c0 = wmma_f32(a, wb0[kk], c0);
            c1 = wmma_f32(a, wb1[kk], c1);
        }

        // Current tile's 4 loads finished once only the 4 newest are in flight
        // (per-wave in-order completion).
        WAIT_ASYNC(4);
        __syncthreads();

        // g = (h @ Whh^T) + gx[t] + b_hh  -> LDS gate buffer
        #pragma unroll
        for (int r = 0; r < 8; ++r) {
            int row = r + half * 8;
            int nA = n0 + lrow, nB = n0 + 16 + lrow;
            Gs[row][nA] = c0[r] + Gxs[cb][row][nA] + bhA;
            Gs[row][nB] = c1[r] + Gxs[cb][row][nB] + bhB;
        }
        __syncthreads();

        // Gate nonlinearities: i|f|g|o slabs of 64 each.
        #pragma unroll
        for (int j = 0; j < 4; ++j) {
            int row = prow[j], h = ph[j];
            float ig = sigf(Gs[row][h]);
            float fg = sigf(Gs[row][64 + h]);
            float gg = tanhf(Gs[row][128 + h]);
            float og = sigf(Gs[row][192 + h]);
            float cv = fg * cst[j] + ig * gg;
            cst[j] = cv;
            float hv = og * tanhf(cv);
            Hs[row][h] = hv;
            hout[hbase + ((size_t)(b0 + row) * T_SZ + t) * HD + h] = hv;
        }
        __syncthreads();
    }
}

// ---------------------------------------------------------------------------
// Kernel 3: emissions = [h_f ; h_b] @ W_out^T + b_out, K padded 17 -> 32.
// Grid: BT/16. Block: 64 (2 waves, one 16-wide N tile each).
// ---------------------------------------------------------------------------
__global__ void k_emissions(const float* __restrict__ hout,
                            const float* __restrict__ W_out,
                            const float* __restrict__ b_out,
                            float* __restrict__ em)
{
    __shared__ __align__(16) float As[16][H3_PAD];   // 16 x 128
    __shared__ __align__(16) float Bs[32][H3_PAD];   // 32 x 128 (rows >=17 zero)

    const int m0  = blockIdx.x * 16;
    const int tid = threadIdx.x;
    const size_t BT = (size_t)B_SZ * T_SZ;

    // A: rows of h_f / h_b (256B-aligned segments) -> B128 async.
    for (int idx = tid; idx < 16 * 2 * 16; idx += 64) {
        int r = idx >> 5, d = (idx >> 4) & 1, cg = idx & 15;
        async_b128(LDSA(&As[r][d * 64 + cg * 4]),
                   hout + (size_t)d * BT * HD + (size_t)(m0 + r) * HD + cg * 4);
    }
    // B: W_out rows (512B-aligned) -> B128 async; zero-fill pad rows.
    for (int idx = tid; idx < K_TAGS * 32; idx += 64) {
        int n = idx >> 5, cg = idx & 31;
        async_b128(LDSA(&Bs[n][cg * 4]), W_out + n * H_DIM + cg * 4);
    }
    for (int idx = tid; idx < (32 - K_TAGS) * H_DIM; idx += 64) {
        int n = K_TAGS + idx / H_DIM, k = idx % H_DIM;
        Bs[n][k] = 0.f;
    }
    WAIT_ASYNC(0);
    __syncthreads();

    const int wv = tid >> 5, lane = tid & 31;
    const int half = lane >> 4, lrow = lane & 15;
    const int n0 = wv * 16;

    v8f c = {};
    #pragma unroll 8
    for (int kk = 0; kk < H_DIM; kk += 4) {   // 32 WMMA steps
        int k = kk + half * 2;
        v2f a, b;
        a[0] = As[lrow][k];       a[1] = As[lrow][k + 1];
        b[0] = Bs[n0 + lrow][k];  b[1] = Bs[n0 + lrow][k + 1];
        c = wmma_f32(a, b, c);
    }

    const int n = n0 + lrow;
    if (n < K_TAGS) {
        float bias = b_out[n];
        #pragma unroll
        for (int r = 0; r < 8; ++r) {
            int row = m0 + r + half * 8;
            em[(size_t)row * K_TAGS + n] = c[r] + bias;
        }
    }
}

// ---------------------------------------------------------------------------
// Kernel 4: CRF gold path score, one wave per batch element.
// ---------------------------------------------------------------------------
__global__ void k_goldscore(const int* __restrict__ sent,
                            const int* __restrict__ tags,
                            const float* __restrict__ em,
                            const float* __restrict__ start_tr,
                            const float* __restrict__ end_tr,
                            const float* __restrict__ trans,
                            float* __restrict__ score)
{
    const int wv = threadIdx.x >> 5, lane = threadIdx.x & 31;
    const int b  = blockIdx.x * 8 + wv;
    const int* s  = sent + (size_t)b * T_SZ;
    const int* tg = tags + (size_t)b * T_SZ;

    float partial = 0.f;
    int cnt = 0;
    for (int t = lane; t < T_SZ; t += 32) {
        int m = (s[t] != 0);
        cnt += m;
        if (t >= 1 && m) {
            int t0 = tg[t - 1], t1 = tg[t];
            partial += em[((size_t)b * T_SZ + t) * K_TAGS + t1]
                     + trans[t0 * K_TAGS + t1];
        }
    }
    #pragma unroll
    for (int off = 16; off; off >>= 1) {
        partial += __shfl_down(partial, off, 32);
        cnt     += __shfl_down(cnt, off, 32);
    }
    if (lane == 0) {
        int t0 = tg[0];
        score[b] = start_tr[t0] + em[((size_t)b * T_SZ) * K_TAGS + t0]
                 + partial + end_tr[tg[cnt - 1]];
    }
}

// ---------------------------------------------------------------------------
// Kernel 5: CRF forward algorithm (logZ), one wave per batch element.
// Lane j (< 17) holds alpha[j]; 17x17 logsumexp per step via __shfl.
// ---------------------------------------------------------------------------
__global__ void k_crf_forward(const int* __restrict__ sent,
                              const float* __restrict__ em,
                              const float* __restrict__ start_tr,
                              const float* __restrict__ end_tr,
                              const float* __restrict__ trans,
                              float* __restrict__ logZ)
{
    __shared__ float tr[K_TAGS][K_TAGS + 1];
    const int tid = threadIdx.x;
    for (int idx = tid; idx < K_TAGS * K_TAGS; idx += 256)
        tr[idx / K_TAGS][idx % K_TAGS] = trans[idx];
    __syncthreads();

    const int wv = tid >> 5, lane = tid & 31;
    const int b  = blockIdx.x * 8 + wv;
    const bool act = lane < K_TAGS;
    const int jj = act ? lane : 0;
    const float NEG = -1e30f;
    const float* eb = em + (size_t)b * T_SZ * K_TAGS;
    const int* s = sent + (size_t)b * T_SZ;

    float alpha = act ? (start_tr[jj] + eb[jj]) : NEG;
    for (int t = 1; t < T_SZ; ++t) {
        float emj = act ? eb[t * K_TAGS + lane] : 0.f;
        int m = (s[t] != 0);
        float vv[K_TAGS];
        float mx = NEG;
        #pragma unroll
        for (int i = 0; i < K_TAGS; ++i) {
            float ai = __shfl(alpha, i, 32);
            vv[i] = ai + tr[i][jj];
            mx = fmaxf(mx, vv[i]);
        }
        float ssum = 0.f;
        #pragma unroll
        for (int i = 0; i < K_TAGS; ++i) ssum += __expf(vv[i] - mx);
        float nxt = mx + __logf(ssum) + emj;
        alpha = (m && act) ? nxt : alpha;
    }
    // wave-wide logsumexp(alpha + end_trans)
    float v = act ? (alpha + end_tr[jj]) : NEG;
    float mx = v;
    #pragma unroll
    for (int off = 16; off; off >>= 1) mx = fmaxf(mx, __shfl_xor(mx, off, 32));
    float e = __expf(v - mx);
    #pragma unroll
    for (int off = 16; off; off >>= 1) e += __shfl_xor(e, off, 32);
    if (lane == 0) logZ[b] = mx + __logf(e);
}

// ---------------------------------------------------------------------------
// Kernel 6: -sum_b (score[b] - logZ[b])
// ---------------------------------------------------------------------------
__global__ void k_finalize(const float* __restrict__ score,
                           const float* __restrict__ logZ,
                           float* __restrict__ out)
{
    __shared__ float red[256];
    const int tid = threadIdx.x;
    red[tid] = score[tid] - logZ[tid];
    __syncthreads();
    for (int sft = 128; sft; sft >>= 1) {
        if (tid < sft) red[tid] += red[tid + sft];
        __syncthreads();
    }
    if (tid == 0) out[0] = -red[0];
}

// ---------------------------------------------------------------------------
extern "C" void kernel_launch(void* const* d_in, const int* in_sizes, int n_in,
                              void* d_out, int out_size, void* d_ws, size_t ws_size,
                              hipStream_t stream)
{
    const int*   sent     = (const int*)d_in[0];
    const int*   tags     = (const int*)d_in[1];
    const float* emb      = (const float*)d_in[2];
    const float* w_ih_f   = (const float*)d_in[3];
    const float* w_hh_f   = (const float*)d_in[4];
    const float* b_ih_f   = (const float*)d_in[5];
    const float* b_hh_f   = (const float*)d_in[6];
    const float* w_ih_b   = (const float*)d_in[7];
    const float* w_hh_b   = (const float*)d_in[8];
    const float* b_ih_b   = (const float*)d_in[9];
    const float* b_hh_b   = (const float*)d_in[10];
    const float* W_out    = (const float*)d_in[11];
    const float* b_out    = (const float*)d_in[12];
    const float* start_tr = (const float*)d_in[13];
    const float* end_tr   = (const float*)d_in[14];
    const float* trans    = (const float*)d_in[15];

    const size_t BT = (size_t)B_SZ * T_SZ;
    float* ws    = (float*)d_ws;
    float* gx    = ws;                      // 2 * BT * 256
    float* hbuf  = gx   + 2 * BT * G4;      // 2 * BT * 64
    float* em    = hbuf + 2 * BT * HD;      // BT * 17
    float* score = em   + BT * K_TAGS;      // 256
    float* logZ  = score + B_SZ;            // 256

    k_input_gemm<<<dim3((unsigned)(BT / 16), 4), 256, 0, stream>>>(
        sent, emb, w_ih_f, b_ih_f, w_ih_b, b_ih_b, gx);
    k_lstm<<<dim3(16, 2), 256, 0, stream>>>(
        gx, w_hh_f, b_hh_f, w_hh_b, b_hh_b, hbuf);
    k_emissions<<<dim3((unsigned)(BT / 16)), 64, 0, stream>>>(
        hbuf, W_out, b_out, em);
    k_goldscore<<<dim3(32), 256, 0, stream>>>(
        sent, tags, em, start_tr, end_tr, trans, score);
    k_crf_forward<<<dim3(32), 256, 0, stream>>>(
        sent, em, start_tr, end_tr, trans, logZ);
    k_finalize<<<dim3(1), 256, 0, stream>>>(score, logZ, (float*)d_out);
}